// GAT_429496729745
// MI455X (gfx1250) — compile-verified
//
#include <hip/hip_runtime.h>
#include <math.h>

// ---------------------------------------------------------------------------
// GAT 2-layer forward for MI455X (gfx1250, wave32).
//  - GEMMs use native fp32 WMMA (v_wmma_f32_16x16x4_f32), fully unrolled
//    K-loop with compile-time addressing (templated K/NCOLS).
//  - Edge softmax: 3 passes with L2 atomics (max / exp+sum / weighted agg).
// ---------------------------------------------------------------------------

typedef __attribute__((ext_vector_type(2))) float v2f;
typedef __attribute__((ext_vector_type(8))) float v8f;

#define NEG_SLOPE 0.2f
#define ENC_NEG_INF 0x007FFFFFu   // enc(-inf)

__device__ __forceinline__ unsigned enc_f32(float f) {
  unsigned u = __float_as_uint(f);
  return (u & 0x80000000u) ? ~u : (u | 0x80000000u);
}
__device__ __forceinline__ float dec_f32(unsigned e) {
  unsigned u = (e & 0x80000000u) ? (e & 0x7FFFFFFFu) : ~e;
  return __uint_as_float(u);
}

__device__ __forceinline__ void get_edge(const int* __restrict__ ei, int E,
                                         int e, int& s, int& d) {
  if (e < E) { s = ei[e]; d = ei[E + e]; }
  else       { s = e - E; d = e - E; }     // self loops appended
}

// ---------------------------------------------------------------- fill -----
__global__ void fill_u32(unsigned* __restrict__ p, unsigned v, long n) {
  long i = (long)blockIdx.x * blockDim.x + threadIdx.x;
  if (i < n) p[i] = v;
}

// ---------------------------------------------------------------- GEMM -----
// C[M,NCOLS] = A[M,K] * B[K,NCOLS], row-major fp32, one 16x16 tile per wave.
// K, NCOLS compile-time: fully unrolled K-loop, immediate load offsets,
// branchless B-column masking (no EXEC churn around the WMMAs).
template <int K, int NCOLS>
__global__ void wmma_gemm_f32(const float* __restrict__ A,
                              const float* __restrict__ B,
                              float* __restrict__ C,
                              int M, int nTiles, int totTiles) {
  int wave = blockIdx.x * (blockDim.x >> 5) + (threadIdx.x >> 5);
  if (wave >= totTiles) return;                 // uniform per wave
  int lane = threadIdx.x & 31;
  int m0 = (wave / nTiles) << 4;
  int n0 = (wave % nTiles) << 4;
  int l15  = lane & 15;
  int kOff = (lane >> 4) << 1;                  // 0 (lanes 0-15) or 2 (16-31)

  // A fragment: lanes l and l+16 both hold row m0+l; K split {0,1} vs {2,3}.
  int rowA = m0 + l15;
  if (rowA >= M) rowA = M - 1;                  // clamp (masked on store)
  const float* Ap = A + (size_t)rowA * K + kOff;

  // B fragment: row k striped across lanes; clamp col in-bounds + 0/1 mask.
  int   colB  = n0 + l15;
  bool  colOK = colB < NCOLS;
  float bmask = colOK ? 1.f : 0.f;
  int   colBc = colOK ? colB : (NCOLS - 1);
  const float* Bp = B + (size_t)kOff * NCOLS + colBc;

  v8f c = {0.f,0.f,0.f,0.f,0.f,0.f,0.f,0.f};
#pragma unroll
  for (int kk = 0; kk < K; kk += 4) {
    v2f a, b;
    a.x = Ap[kk];                               // contiguous pair -> b64 load
    a.y = Ap[kk + 1];
    b.x = bmask * Bp[(size_t)kk * NCOLS];       // immediate offsets
    b.y = bmask * Bp[(size_t)(kk + 1) * NCOLS];
    // D = A*B + C (neg_a, A, neg_b, B, c_mod, C, reuseA, reuseB)
    c = __builtin_amdgcn_wmma_f32_16x16x4_f32(false, a, false, b,
                                              (short)0, c, false, false);
  }

  if (colOK) {
    int rowBase = m0 + ((lane >> 4) << 3);      // VGPR j -> rows j and j+8
    float* Cp = C + (size_t)rowBase * NCOLS + colB;
#pragma unroll
    for (int j = 0; j < 8; ++j)
      if (rowBase + j < M) Cp[(size_t)j * NCOLS] = c[j];
  }
}

// ------------------------------------------------- per-node attn coeffs ----
// a_s[n,h] = dot(h[n,h,:], att_src[h,:]);  a_d likewise.
__global__ void attn_coeff(const float* __restrict__ hmat,
                           const float* __restrict__ ws,
                           const float* __restrict__ wd,
                           float* __restrict__ a_s, float* __restrict__ a_d,
                           int N, int H, int C) {
  int t = blockIdx.x * blockDim.x + threadIdx.x;
  if (t >= N * H) return;
  int n = t / H, h = t % H;
  const float* hp = hmat + (long)n * H * C + (long)h * C;
  const float* as = ws + h * C;
  const float* ad = wd + h * C;
  float s = 0.f, d = 0.f;
  for (int c = 0; c < C; ++c) {
    float v = hp[c];
    s = fmaf(v, as[c], s);
    d = fmaf(v, ad[c], d);
  }
  a_s[t] = s;
  a_d[t] = d;
}

// -------------------------------------------------- edge pass 1: max -------
__global__ void edge_max(const int* __restrict__ ei, int E, int H,
                         const float* __restrict__ a_s,
                         const float* __restrict__ a_d,
                         unsigned* __restrict__ menc, long tot) {
  long t = (long)blockIdx.x * blockDim.x + threadIdx.x;
  if (t >= tot) return;
  int e = (int)(t / H), h = (int)(t % H);
  int s, d; get_edge(ei, E, e, s, d);
  float logit = a_s[(long)s * H + h] + a_d[(long)d * H + h];
  logit = logit >= 0.f ? logit : NEG_SLOPE * logit;
  atomicMax(&menc[(long)d * H + h], enc_f32(logit));
}

// ---------------------------------------- edge pass 2: exp + denominator ---
__global__ void edge_exp(const int* __restrict__ ei, int E, int H,
                         const float* __restrict__ a_s,
                         const float* __restrict__ a_d,
                         const unsigned* __restrict__ menc,
                         float* __restrict__ exbuf,
                         float* __restrict__ den, long tot) {
  long t = (long)blockIdx.x * blockDim.x + threadIdx.x;
  if (t >= tot) return;
  int e = (int)(t / H), h = (int)(t % H);
  int s, d; get_edge(ei, E, e, s, d);
  long dh = (long)d * H + h;
  float logit = a_s[(long)s * H + h] + a_d[dh];
  logit = logit >= 0.f ? logit : NEG_SLOPE * logit;
  float m = dec_f32(menc[dh]);
  if (!isfinite(m)) m = 0.f;                    // ref: where(isfinite(m), m, 0)
  float exv = __expf(logit - m);
  exbuf[t] = exv;
  atomicAdd(&den[dh], exv);
}

// --------------------------------------- edge pass 3: weighted aggregate ---
// One wave32 per (edge, head): lanes cover the C channels, coalesced atomics.
__global__ void edge_agg(const int* __restrict__ ei, int E, int H, int C,
                         const float* __restrict__ hmat,
                         const float* __restrict__ exbuf,
                         const float* __restrict__ den,
                         float* __restrict__ outm, long totWaves) {
  long w = ((long)blockIdx.x * blockDim.x + threadIdx.x) >> 5;
  if (w >= totWaves) return;                    // uniform per wave
  int lane = threadIdx.x & 31;
  int e = (int)(w / H), h = (int)(w % H);
  int s, d; get_edge(ei, E, e, s, d);
  float alpha = exbuf[w] / (den[(long)d * H + h] + 1e-16f);
  const float* hs = hmat + (long)s * H * C + (long)h * C;
  float*       od = outm + (long)d * H * C + (long)h * C;
  for (int c = lane; c < C; c += 32)
    atomicAdd(&od[c], alpha * hs[c]);
}

// ------------------------------------------------------- bias / act --------
__global__ void bias_elu_inplace(float* __restrict__ x,
                                 const float* __restrict__ b, int F, long n) {
  long i = (long)blockIdx.x * blockDim.x + threadIdx.x;
  if (i >= n) return;
  float v = x[i] + b[i % F];
  x[i] = v > 0.f ? v : expm1f(v);
}
__global__ void bias_add_inplace(float* __restrict__ x,
                                 const float* __restrict__ b, int F, long n) {
  long i = (long)blockIdx.x * blockDim.x + threadIdx.x;
  if (i >= n) return;
  x[i] = x[i] + b[i % F];
}

// ---------------------------------------------------------------------------
static inline long div_up(long a, long b) { return (a + b - 1) / b; }

extern "C" void kernel_launch(void* const* d_in, const int* in_sizes, int n_in,
                              void* d_out, int out_size, void* d_ws, size_t ws_size,
                              hipStream_t stream) {
  const float* x     = (const float*)d_in[0];   // [N,128]
  const int*   ei    = (const int*)  d_in[1];   // [2,E]
  const float* W1    = (const float*)d_in[2];   // [128,128]
  const float* asrc1 = (const float*)d_in[3];   // [2,64]
  const float* adst1 = (const float*)d_in[4];   // [2,64]
  const float* b1    = (const float*)d_in[5];   // [128]
  const float* W2    = (const float*)d_in[6];   // [128,40]
  const float* asrc2 = (const float*)d_in[7];   // [1,40]
  const float* adst2 = (const float*)d_in[8];   // [1,40]
  const float* b2    = (const float*)d_in[9];   // [40]
  float* out = (float*)d_out;                   // [N,40]

  const int IN_CH = 128, HID = 64, HEADS = 2, OUT_CH = 40;
  const int F1 = HEADS * HID;                   // 128
  const int N = in_sizes[0] / IN_CH;
  const int E = in_sizes[1] / 2;
  const int Etot = E + N;                       // with self loops

  // ---- workspace bump allocator (256B aligned) ----
  char* wsp = (char*)d_ws;
  size_t off = 0;
  auto alloc = [&](size_t bytes) -> void* {
    void* p = wsp + off;
    off += (bytes + 255) & ~(size_t)255;
    return p;
  };
  float*    h1    = (float*)   alloc((size_t)N * F1 * 4);     // layer1 features
  float*    agg1  = (float*)   alloc((size_t)N * F1 * 4);     // layer1 aggregate / x2
  float*    as1   = (float*)   alloc((size_t)N * HEADS * 4);
  float*    ad1   = (float*)   alloc((size_t)N * HEADS * 4);
  unsigned* menc1 = (unsigned*)alloc((size_t)N * HEADS * 4);
  float*    den1  = (float*)   alloc((size_t)N * HEADS * 4);
  float*    ex1   = (float*)   alloc((size_t)Etot * HEADS * 4);
  float*    g2    = (float*)   alloc((size_t)N * OUT_CH * 4); // layer2 features
  float*    as2   = (float*)   alloc((size_t)N * 4);
  float*    ad2   = (float*)   alloc((size_t)N * 4);
  unsigned* menc2 = (unsigned*)alloc((size_t)N * 4);
  float*    den2  = (float*)   alloc((size_t)N * 4);
  float*    ex2   = (float*)   alloc((size_t)Etot * 4);
  (void)ws_size; (void)n_in; (void)out_size;

  const int BLK = 256;

  // ================= Layer 1 =================
  {
    int nT = F1 / 16, tot = (int)div_up(N, 16) * nT;
    wmma_gemm_f32<128, 128><<<(int)div_up(tot, BLK / 32), BLK, 0, stream>>>(
        x, W1, h1, N, nT, tot);
  }
  attn_coeff<<<(int)div_up((long)N * HEADS, BLK), BLK, 0, stream>>>(
      h1, asrc1, adst1, as1, ad1, N, HEADS, HID);

  fill_u32<<<(int)div_up((long)N * HEADS, BLK), BLK, 0, stream>>>(
      menc1, ENC_NEG_INF, (long)N * HEADS);
  fill_u32<<<(int)div_up((long)N * HEADS, BLK), BLK, 0, stream>>>(
      (unsigned*)den1, 0u, (long)N * HEADS);
  fill_u32<<<(int)div_up((long)N * F1, BLK), BLK, 0, stream>>>(
      (unsigned*)agg1, 0u, (long)N * F1);

  {
    long tot = (long)Etot * HEADS;
    edge_max<<<(int)div_up(tot, BLK), BLK, 0, stream>>>(ei, E, HEADS, as1, ad1, menc1, tot);
    edge_exp<<<(int)div_up(tot, BLK), BLK, 0, stream>>>(ei, E, HEADS, as1, ad1, menc1, ex1, den1, tot);
    edge_agg<<<(int)div_up(tot * 32, BLK), BLK, 0, stream>>>(ei, E, HEADS, HID, h1, ex1, den1, agg1, tot);
  }
  bias_elu_inplace<<<(int)div_up((long)N * F1, BLK), BLK, 0, stream>>>(
      agg1, b1, F1, (long)N * F1);              // agg1 is now x2 (ELU'd)

  // ================= Layer 2 =================
  {
    int nT = (OUT_CH + 15) / 16, tot = (int)div_up(N, 16) * nT;
    wmma_gemm_f32<128, 40><<<(int)div_up(tot, BLK / 32), BLK, 0, stream>>>(
        agg1, W2, g2, N, nT, tot);
  }
  attn_coeff<<<(int)div_up((long)N, BLK), BLK, 0, stream>>>(
      g2, asrc2, adst2, as2, ad2, N, 1, OUT_CH);

  fill_u32<<<(int)div_up((long)N, BLK), BLK, 0, stream>>>(menc2, ENC_NEG_INF, (long)N);
  fill_u32<<<(int)div_up((long)N, BLK), BLK, 0, stream>>>((unsigned*)den2, 0u, (long)N);
  fill_u32<<<(int)div_up((long)N * OUT_CH, BLK), BLK, 0, stream>>>(
      (unsigned*)out, 0u, (long)N * OUT_CH);

  {
    long tot = (long)Etot;                      // H = 1
    edge_max<<<(int)div_up(tot, BLK), BLK, 0, stream>>>(ei, E, 1, as2, ad2, menc2, tot);
    edge_exp<<<(int)div_up(tot, BLK), BLK, 0, stream>>>(ei, E, 1, as2, ad2, menc2, ex2, den2, tot);
    edge_agg<<<(int)div_up(tot * 32, BLK), BLK, 0, stream>>>(ei, E, 1, OUT_CH, g2, ex2, den2, out, tot);
  }
  bias_add_inplace<<<(int)div_up((long)N * OUT_CH, BLK), BLK, 0, stream>>>(
      out, b2, OUT_CH, (long)N * OUT_CH);
}